// Doc3DDataAug_44220983280425
// MI455X (gfx1250) — compile-verified
//
#include <hip/hip_runtime.h>
#include <cstdint>
#include <cstddef>

#define BN    64
#define HH    448
#define WW    448
#define HW    200704          // 448*448
#define QPR   112             // quads per row (448/4)
#define OUTD  256
#define OUT2  65536           // 256*256
#define HBW   486             // 448 + 2*(20-1)
#define HBWB  236196          // 486*486
#define XSIZE (BN*3*OUT2)     // 12,582,912 floats

// ---------------- workspace layout (bytes) ----------------
#define WS_BBOX 0                         // 64*4 ints   = 1024 B
#define WS_MEAN 1024                      // 64 floats   = 256 B
#define WS_PART 4096                      // 64*256 f32  = 65536 B
#define WS_MASK (4096 + 65536)            // 64*200704 u8 = 12,845,056 B

__device__ __forceinline__ float clip01(float x) {
  return fminf(fmaxf(x, 0.0f), 1.0f);
}

// ---------------------------------------------------------------------------
// Kernel 1: mask bytes + per-sample bounding box (64 blocks x 256 threads).
// Quad-vectorized: 3x global_load_b128 + 1x packed b32 mask store per 4 px.
// ---------------------------------------------------------------------------
__global__ void k_mask_bbox(const float* __restrict__ wc,
                            unsigned char* __restrict__ maskb,
                            int* __restrict__ bbox) {
  const int b   = blockIdx.x;
  const int tid = threadIdx.x;
  const float* w0 = wc + (size_t)b * 3 * HW;
  const float* w1 = w0 + HW;
  const float* w2 = w1 + HW;
  uchar4* mb4 = (uchar4*)(maskb + (size_t)b * HW);

  int miny = HH, maxy = -1, minx = WW, maxx = -1;
  const int NQ = HW / 4;                  // 50176 quads; quads never cross rows
  for (int q = tid; q < NQ; q += 256) {
    __builtin_prefetch(w0 + q * 4 + 8192, 0, 1);   // global_prefetch_b8
    const float4 a = *(const float4*)(w0 + q * 4);
    const float4 c = *(const float4*)(w1 + q * 4);
    const float4 d = *(const float4*)(w2 + q * 4);
    const bool m0 = (a.x != 0.0f) && (c.x != 0.0f) && (d.x != 0.0f);
    const bool m1 = (a.y != 0.0f) && (c.y != 0.0f) && (d.y != 0.0f);
    const bool m2 = (a.z != 0.0f) && (c.z != 0.0f) && (d.z != 0.0f);
    const bool m3 = (a.w != 0.0f) && (c.w != 0.0f) && (d.w != 0.0f);
    mb4[q] = make_uchar4(m0 ? 1 : 0, m1 ? 1 : 0, m2 ? 1 : 0, m3 ? 1 : 0);
    if (m0 | m1 | m2 | m3) {
      const int y  = q / QPR;
      const int xb = (q - y * QPR) * 4;
      miny = min(miny, y); maxy = max(maxy, y);
      const int lo = m0 ? 0 : (m1 ? 1 : (m2 ? 2 : 3));
      const int hi = m3 ? 3 : (m2 ? 2 : (m1 ? 1 : 0));
      minx = min(minx, xb + lo); maxx = max(maxx, xb + hi);
    }
  }
  __shared__ int s[4];
  if (tid == 0) { s[0] = HH; s[1] = -1; s[2] = WW; s[3] = -1; }
  __syncthreads();
  atomicMin(&s[0], miny); atomicMax(&s[1], maxy);
  atomicMin(&s[2], minx); atomicMax(&s[3], maxx);
  __syncthreads();
  if (tid < 4) bbox[b * 4 + tid] = s[tid];
}

// ---------------------------------------------------------------------------
// Kernel 2: composite + bilinear resize + brightness; per-row gray partials.
// One block = one output row: blockIdx.x = b*256 + py, threadIdx.x = px.
// ---------------------------------------------------------------------------
__global__ void k_composite(const float* __restrict__ img,
                            const unsigned char* __restrict__ maskb,
                            const float* __restrict__ bg,
                            const float* __restrict__ rcol,
                            const float* __restrict__ bright,
                            const int* __restrict__ cpar,
                            const int* __restrict__ bbox,
                            float* __restrict__ outx,
                            float* __restrict__ partial) {
  const int b  = blockIdx.x >> 8;
  const int py = blockIdx.x & 255;
  const int px = threadIdx.x;

  const int miny = bbox[b * 4 + 0], maxy = bbox[b * 4 + 1];
  const int minx = bbox[b * 4 + 2], maxx = bbox[b * 4 + 3];
  const int c0 = cpar[b * 5 + 0], c1 = cpar[b * 5 + 1];
  const int c2 = cpar[b * 5 + 2], c3 = cpar[b * 5 + 3];
  const int c4 = cpar[b * 5 + 4];
  const int Hp = maxy - miny + 1 + c2 + c3;
  const int Wp = maxx - minx + 1 + c0 + c1;

  // src_coords (matches reference: weight computed before clamping i0)
  float sy = ((float)py + 0.5f) * ((float)Hp / (float)OUTD) - 0.5f;
  sy = fmaxf(sy, 0.0f);
  int   y0 = (int)floorf(sy);
  float wy = sy - (float)y0;
  y0 = min(y0, Hp - 1);
  int   y1 = min(y0 + 1, Hp - 1);

  float sx = ((float)px + 0.5f) * ((float)Wp / (float)OUTD) - 0.5f;
  sx = fmaxf(sx, 0.0f);
  int   x0 = (int)floorf(sx);
  float wx = sx - (float)x0;
  x0 = min(x0, Wp - 1);
  int   x1 = min(x0 + 1, Wp - 1);

  const float* ib  = img + (size_t)b * 3 * HW;
  const unsigned char* mb = maskb + (size_t)b * HW;
  const float* bgb = bg + (size_t)b * 3 * HBWB;
  const float colr = rcol[b * 3 + 0];
  const float colg = rcol[b * 3 + 1];
  const float colb = rcol[b * 3 + 2];
  const bool useBg = (c4 > 2);

  const int   ys[2]  = { y0, y1 };
  const int   xs[2]  = { x0, x1 };
  const float wys[2] = { 1.0f - wy, wy };
  const float wxs[2] = { 1.0f - wx, wx };

  float accr = 0.0f, accg = 0.0f, accb = 0.0f;
#pragma unroll
  for (int iy = 0; iy < 2; ++iy) {
#pragma unroll
    for (int ix = 0; ix < 2; ++ix) {
      const int yS = ys[iy], xS = xs[ix];
      const int oy = yS - c2 + miny;
      const int ox = xS - c0 + minx;
      const bool valid = (oy >= miny) && (oy <= maxy) && (ox >= minx) && (ox <= maxx);
      const int oyc = min(max(oy, 0), HH - 1);
      const int oxc = min(max(ox, 0), WW - 1);
      const float m = valid ? (float)mb[oyc * WW + oxc] : 0.0f;
      const float one_m = 1.0f - m;
      const int byc = min(max(yS, 0), HBW - 1);
      const int bxc = min(max(xS, 0), HBW - 1);
      const int iidx = oyc * WW + oxc;
      const int bidx = byc * HBW + bxc;
      const float bg0 = useBg ? bgb[bidx]             : colr;
      const float bg1 = useBg ? bgb[HBWB + bidx]      : colg;
      const float bg2 = useBg ? bgb[2 * HBWB + bidx]  : colb;
      const float w = wys[iy] * wxs[ix];
      accr += w * (ib[iidx]          * m + bg0 * one_m);
      accg += w * (ib[HW + iidx]     * m + bg1 * one_m);
      accb += w * (ib[2 * HW + iidx] * m + bg2 * one_m);
    }
  }

  const float brt = bright[b];
  const float r  = clip01(accr * brt);
  const float g  = clip01(accg * brt);
  const float bl = clip01(accb * brt);

  const size_t o = (size_t)b * 3 * OUT2 + (size_t)py * OUTD + px;
  outx[o]            = r;
  outx[o + OUT2]     = g;
  outx[o + 2 * OUT2] = bl;

  // deterministic fixed-order block reduction of gray
  __shared__ float sd[256];
  sd[px] = 0.299f * r + 0.587f * g + 0.114f * bl;
  __syncthreads();
#pragma unroll
  for (int s = 128; s > 0; s >>= 1) {
    if (px < s) sd[px] += sd[px + s];
    __syncthreads();
  }
  if (px == 0) partial[blockIdx.x] = sd[0];
}

// ---------------------------------------------------------------------------
// Kernel 3: per-sample mean via CDNA5 async-to-LDS load + fixed tree reduce.
// ---------------------------------------------------------------------------
__global__ void k_mean(const float* __restrict__ partial,
                       float* __restrict__ meanv) {
  const int b = blockIdx.x;
  const int t = threadIdx.x;
  __shared__ float sp[256];

  const float* gp = partial + b * 256 + t;
  unsigned ldsoff = (unsigned)(size_t)(&sp[t]);   // low 32 bits = LDS byte offset
  asm volatile("global_load_async_to_lds_b32 %0, %1, off"
               :: "v"(ldsoff), "v"(gp) : "memory");
  asm volatile("s_wait_asynccnt 0" ::: "memory");
  __syncthreads();

#pragma unroll
  for (int s = 128; s > 0; s >>= 1) {
    if (t < s) sp[t] += sp[t + s];
    __syncthreads();
  }
  if (t == 0) meanv[b] = sp[0] * (1.0f / 65536.0f);
}

// ---------------------------------------------------------------------------
// Per-pixel color pipeline: contrast + saturation + HSV hue shift.
// ---------------------------------------------------------------------------
__device__ __forceinline__ void color_one(float& r, float& g, float& bl,
                                          float mean, float cf, float sf,
                                          float hu) {
  // contrast
  r  = clip01(cf * r  + (1.0f - cf) * mean);
  g  = clip01(cf * g  + (1.0f - cf) * mean);
  bl = clip01(cf * bl + (1.0f - cf) * mean);

  // saturation
  const float gr = 0.299f * r + 0.587f * g + 0.114f * bl;
  r  = clip01(sf * r  + (1.0f - sf) * gr);
  g  = clip01(sf * g  + (1.0f - sf) * gr);
  bl = clip01(sf * bl + (1.0f - sf) * gr);

  // rgb -> hsv
  const float eps = 1e-8f;
  const float mx = fmaxf(r, fmaxf(g, bl));
  const float mn = fminf(r, fminf(g, bl));
  const float d  = mx - mn;
  float hr = fmodf((g - bl) / (d + eps), 6.0f);
  if (hr < 0.0f) hr += 6.0f;                    // python-style mod
  const float hg = (bl - r) / (d + eps) + 2.0f;
  const float hb = (r - g)  / (d + eps) + 4.0f;
  float h = ((mx == r) ? hr : ((mx == g) ? hg : hb)) * (1.0f / 6.0f);
  if (d <= eps) h = 0.0f;
  const float s = d / (mx + eps);
  const float v = mx;

  // hue shift, python-style mod into [0,1)
  h += hu;
  h -= floorf(h);

  // hsv -> rgb
  const float h6 = (h - floorf(h)) * 6.0f;
  const float fi = floorf(h6);
  const float f  = h6 - fi;
  const float p  = v * (1.0f - s);
  const float q  = v * (1.0f - f * s);
  const float tt = v * (1.0f - (1.0f - f) * s);
  const int i = ((int)fi) % 6;
  float rr, gg, bb;
  switch (i) {
    case 0:  rr = v;  gg = tt; bb = p;  break;
    case 1:  rr = q;  gg = v;  bb = p;  break;
    case 2:  rr = p;  gg = v;  bb = tt; break;
    case 3:  rr = p;  gg = q;  bb = v;  break;
    case 4:  rr = tt; gg = p;  bb = v;  break;
    default: rr = v;  gg = p;  bb = q;  break;
  }
  r  = clip01(rr);
  g  = clip01(gg);
  bl = clip01(bb);
}

// ---------------------------------------------------------------------------
// Kernel 4: color jitter in place on d_out, 4 pixels/thread (b128 ld/st).
// gid covers BN * OUT2/4 quads; 16384 quads per sample.
// ---------------------------------------------------------------------------
__global__ void k_color(float* __restrict__ outx,
                        const float* __restrict__ meanv,
                        const float* __restrict__ con,
                        const float* __restrict__ sat,
                        const float* __restrict__ huep) {
  const int gid = blockIdx.x * 256 + threadIdx.x;
  const int b   = gid >> 14;                    // / (OUT2/4)
  const int q   = gid & 16383;
  const size_t o = (size_t)b * 3 * OUT2 + (size_t)q * 4;

  float4 R  = *(float4*)(outx + o);
  float4 G  = *(float4*)(outx + o + OUT2);
  float4 Bl = *(float4*)(outx + o + 2 * OUT2);

  const float mean = meanv[b];
  const float cf = con[b];
  const float sf = sat[b];
  const float hu = huep[b];

  color_one(R.x, G.x, Bl.x, mean, cf, sf, hu);
  color_one(R.y, G.y, Bl.y, mean, cf, sf, hu);
  color_one(R.z, G.z, Bl.z, mean, cf, sf, hu);
  color_one(R.w, G.w, Bl.w, mean, cf, sf, hu);

  *(float4*)(outx + o)            = R;
  *(float4*)(outx + o + OUT2)     = G;
  *(float4*)(outx + o + 2 * OUT2) = Bl;
}

// ---------------------------------------------------------------------------
// Kernel 5: bm remap, float4-vectorized (b128 loads/stores).
// ---------------------------------------------------------------------------
__global__ void k_bm(const float* __restrict__ bm,
                     const int* __restrict__ bbox,
                     const int* __restrict__ cpar,
                     float* __restrict__ outbm) {
  const int gid = blockIdx.x * 256 + threadIdx.x;
  const int b  = gid / (HW / 4);
  const int p  = (gid - b * (HW / 4)) * 4;

  const int miny = bbox[b * 4 + 0], maxy = bbox[b * 4 + 1];
  const int minx = bbox[b * 4 + 2], maxx = bbox[b * 4 + 3];
  const int c0 = cpar[b * 5 + 0], c1 = cpar[b * 5 + 1];
  const int c2 = cpar[b * 5 + 2], c3 = cpar[b * 5 + 3];
  const float Wp = (float)(maxx - minx + 1 + c0 + c1);
  const float Hp = (float)(maxy - miny + 1 + c2 + c3);
  const float ax = (float)c0 - (float)minx;
  const float ay = (float)c2 - (float)miny;

  const float4 vx = *(const float4*)(bm + (size_t)b * 2 * HW + p);
  const float4 vy = *(const float4*)(bm + (size_t)b * 2 * HW + HW + p);
  float4 ox, oy;
  ox.x = (vx.x + ax) / Wp * 2.0f - 1.0f;
  ox.y = (vx.y + ax) / Wp * 2.0f - 1.0f;
  ox.z = (vx.z + ax) / Wp * 2.0f - 1.0f;
  ox.w = (vx.w + ax) / Wp * 2.0f - 1.0f;
  oy.x = (vy.x + ay) / Hp * 2.0f - 1.0f;
  oy.y = (vy.y + ay) / Hp * 2.0f - 1.0f;
  oy.z = (vy.z + ay) / Hp * 2.0f - 1.0f;
  oy.w = (vy.w + ay) / Hp * 2.0f - 1.0f;

  float* ob = outbm + (size_t)b * 2 * HW;
  *(float4*)(ob + p)      = ox;
  *(float4*)(ob + HW + p) = oy;
}

// ---------------------------------------------------------------------------
extern "C" void kernel_launch(void* const* d_in, const int* in_sizes, int n_in,
                              void* d_out, int out_size, void* d_ws, size_t ws_size,
                              hipStream_t stream) {
  (void)in_sizes; (void)n_in; (void)out_size; (void)ws_size;
  const float* img    = (const float*)d_in[0];
  const float* wc     = (const float*)d_in[1];
  const float* bm     = (const float*)d_in[2];
  const float* bg     = (const float*)d_in[3];
  const float* rcol   = (const float*)d_in[4];
  const float* bright = (const float*)d_in[5];
  const float* con    = (const float*)d_in[6];
  const float* sat    = (const float*)d_in[7];
  const float* hue    = (const float*)d_in[8];
  const int*   cpar   = (const int*)d_in[9];

  char* ws = (char*)d_ws;
  int*           bbox    = (int*)(ws + WS_BBOX);
  float*         meanv   = (float*)(ws + WS_MEAN);
  float*         partial = (float*)(ws + WS_PART);
  unsigned char* maskb   = (unsigned char*)(ws + WS_MASK);

  float* outx  = (float*)d_out;
  float* outbm = outx + XSIZE;

  k_mask_bbox<<<BN, 256, 0, stream>>>(wc, maskb, bbox);
  k_composite<<<BN * 256, 256, 0, stream>>>(img, maskb, bg, rcol, bright, cpar,
                                            bbox, outx, partial);
  k_mean<<<BN, 256, 0, stream>>>(partial, meanv);
  k_color<<<(BN * (OUT2 / 4)) / 256, 256, 0, stream>>>(outx, meanv, con, sat, hue);
  k_bm<<<(BN * (HW / 4)) / 256, 256, 0, stream>>>(bm, bbox, cpar, outbm);
}